// GAT_48318382080262
// MI455X (gfx1250) — compile-verified
//
#include <hip/hip_runtime.h>
#include <math.h>

#define NN   50000
#define EE   640000
#define DD   128
#define PED  37
#define PEDP 40      // PE_DIM padded to multiple of 4 for WMMA K-steps
#define NT   9
#define NL   3

typedef __attribute__((ext_vector_type(2))) float v2f;
typedef __attribute__((ext_vector_type(8))) float v8f;

// monotone int encoding of IEEE float for atomicMax-based float max
__device__ __forceinline__ int f2ord(float f) {
  int i = __float_as_int(f);
  return (i >= 0) ? i : (i ^ 0x7fffffff);
}
__device__ __forceinline__ float ord2f(int i) {
  return __int_as_float((i >= 0) ? i : (i ^ 0x7fffffff));
}

// ---------------- init / copy / pad ----------------

__global__ void k_copy_x(const float* __restrict__ X, float* __restrict__ xbuf) {
  int i = blockIdx.x * 256 + threadIdx.x;
  xbuf[i] = X[i];
}

__global__ void k_pad_pe(const float* __restrict__ PE, float* __restrict__ PEp) {
  int i = blockIdx.x * 256 + threadIdx.x;
  if (i >= NN * PEDP) return;
  int r = i / PEDP, c = i - r * PEDP;
  PEp[i] = (c < PED) ? PE[r * PED + c] : 0.0f;
}

__global__ void k_pad_w(const float* __restrict__ W, float* __restrict__ Wp) {
  int i = blockIdx.x * 256 + threadIdx.x;
  if (i >= PEDP * DD) return;
  int k = i >> 7;
  Wp[i] = (k < PED) ? W[i] : 0.0f;
}

__global__ void k_init(float* __restrict__ outb, int* __restrict__ amax,
                       float* __restrict__ denom, float* __restrict__ ssum,
                       float* __restrict__ ssq) {
  int i = blockIdx.x * 256 + threadIdx.x;
  outb[i] = 0.0f;                       // N*D elements
  if (i < NN) { amax[i] = f2ord(-INFINITY); denom[i] = 0.0f; }
  if (i < DD) { ssum[i] = 0.0f; ssq[i] = 0.0f; }
}

// ---------------- WMMA GEMM: C[N x 128] = A[N x K] @ B[K x 128] ----------------
// One wave computes one 16x16 tile with V_WMMA_F32_16X16X4_F32.
// A layout (16x4 f32): lanes 0-15 row M, {K=k0,k0+1}; lanes 16-31 row M, {K=k0+2,k0+3}
// B layout (4x16 f32): lanes 0-15 col N, {K=k0,k0+1}; lanes 16-31 col N, {K=k0+2,k0+3}
// C layout: VGPR r -> row = 16*tm + r + 8*(lane>>4), col = 16*tn + (lane&15)
template <int K, bool PE_EPI>
__global__ void k_gemm(const float* __restrict__ A, const float* __restrict__ B,
                       const float* __restrict__ X, const float* __restrict__ bb,
                       float* __restrict__ C) {
  int wave = threadIdx.x >> 5;
  int lane = threadIdx.x & 31;
  int t  = blockIdx.x * 8 + wave;   // 3125*8 tiles total (N=50000 -> 3125 row tiles)
  int tn = t & 7;
  int tm = t >> 3;
  int mlo = lane & 15;
  int hi  = lane >> 4;              // 0 or 1
  const float* arow = A + (size_t)(tm * 16 + mlo) * K;
  int ncol = tn * 16 + mlo;

  v8f c = {};
  #pragma unroll 4
  for (int k0 = 0; k0 < K; k0 += 4) {
    v2f a = *(const v2f*)(arow + k0 + hi * 2);
    v2f b;
    b.x = B[(size_t)(k0 + hi * 2    ) * DD + ncol];
    b.y = B[(size_t)(k0 + hi * 2 + 1) * DD + ncol];
    c = __builtin_amdgcn_wmma_f32_16x16x4_f32(false, a, false, b, (short)0, c,
                                              false, false);
  }
  #pragma unroll
  for (int r = 0; r < 8; ++r) {
    int row = tm * 16 + r + hi * 8;
    float v = c[r];
    if (PE_EPI) v += X[(size_t)row * DD + ncol] + bb[ncol];
    C[(size_t)row * DD + ncol] = v;
  }
}

// ---------------- per-node attention scores (wave per node) ----------------

__global__ void k_scores(const float* __restrict__ h, const float* __restrict__ al,
                         const float* __restrict__ ar, float* __restrict__ sl,
                         float* __restrict__ sr) {
  int wave = threadIdx.x >> 5, lane = threadIdx.x & 31;
  int n = blockIdx.x * 8 + wave;
  const float* row = h + (size_t)n * DD;
  float accl = 0.0f, accr = 0.0f;
  #pragma unroll
  for (int j = 0; j < 4; ++j) {
    int d = lane + 32 * j;
    float v = row[d];
    accl += v * al[d];
    accr += v * ar[d];
  }
  #pragma unroll
  for (int off = 16; off; off >>= 1) {
    accl += __shfl_xor(accl, off, 32);
    accr += __shfl_xor(accr, off, 32);
  }
  if (lane == 0) { sl[n] = accl; sr[n] = accr; }
}

// edge-type score lookup table: xe[t] = dot(edge_emb[t], attn_e)
__global__ void k_xe(const float* __restrict__ ee, const float* __restrict__ ae,
                     float* __restrict__ xe) {
  int wave = threadIdx.x >> 5, lane = threadIdx.x & 31;
  if (wave >= NT) return;
  float acc = 0.0f;
  #pragma unroll
  for (int j = 0; j < 4; ++j) {
    int d = lane + 32 * j;
    acc += ee[wave * DD + d] * ae[d];
  }
  #pragma unroll
  for (int off = 16; off; off >>= 1) acc += __shfl_xor(acc, off, 32);
  if (lane == 0) xe[wave] = acc;
}

// ---------------- edge passes ----------------

__global__ void k_edge_max(const float* __restrict__ sl, const float* __restrict__ sr,
                           const float* __restrict__ xe, const int* __restrict__ src,
                           const int* __restrict__ dst, const int* __restrict__ attr,
                           float* __restrict__ alpha, int* __restrict__ amax) {
  int e = blockIdx.x * 256 + threadIdx.x;
  int s = src[e], d = dst[e];
  float a = sl[s] + sr[d] + xe[attr[e]];
  a = (a > 0.0f) ? a : 0.2f * a;       // leaky_relu(0.2)
  alpha[e] = a;
  atomicMax(&amax[d], f2ord(a));
}

__global__ void k_edge_exp(const int* __restrict__ amax, const int* __restrict__ dst,
                           float* __restrict__ alpha, float* __restrict__ denom) {
  int e = blockIdx.x * 256 + threadIdx.x;
  int d = dst[e];
  float ex = expf(alpha[e] - ord2f(amax[d]));
  alpha[e] = ex;
  atomicAdd(&denom[d], ex);
}

// weighted scatter-add aggregation: wave per edge, 4 columns per lane
__global__ void k_aggregate(const float* __restrict__ h, const float* __restrict__ alpha,
                            const float* __restrict__ denom, const int* __restrict__ src,
                            const int* __restrict__ dst, float* __restrict__ outb) {
  int wave = threadIdx.x >> 5, lane = threadIdx.x & 31;
  int e = blockIdx.x * 8 + wave;
  int s = src[e], d = dst[e];
  float coeff = alpha[e] / denom[d];
  float4 v = ((const float4*)(h + (size_t)s * DD))[lane];
  float* op = outb + (size_t)d * DD + lane * 4;
  atomicAdd(op + 0, coeff * v.x);
  atomicAdd(op + 1, coeff * v.y);
  atomicAdd(op + 2, coeff * v.z);
  atomicAdd(op + 3, coeff * v.w);
}

// ---------------- BN + residual ----------------

__global__ void k_bias_stats(float* __restrict__ outb, const float* __restrict__ bias,
                             float* __restrict__ ssum, float* __restrict__ ssq) {
  int d = threadIdx.x;                  // 128 threads, one column each
  int r0 = blockIdx.x * 128;
  float b = bias[d];
  float s = 0.0f, s2 = 0.0f;
  for (int r = r0; r < r0 + 128 && r < NN; ++r) {
    float v = outb[(size_t)r * DD + d] + b;
    outb[(size_t)r * DD + d] = v;
    s += v; s2 += v * v;
  }
  atomicAdd(&ssum[d], s);
  atomicAdd(&ssq[d], s2);
}

__global__ void k_bn_residual(const float* __restrict__ outb, const float* __restrict__ ssum,
                              const float* __restrict__ ssq, const float* __restrict__ gam,
                              const float* __restrict__ bet, float* __restrict__ xbuf) {
  int i = blockIdx.x * 256 + threadIdx.x;
  int d = i & (DD - 1);
  float mu  = ssum[d] * (1.0f / NN);
  float var = ssq[d] * (1.0f / NN) - mu * mu;
  float inv = rsqrtf(var + 1e-5f);
  xbuf[i] = (outb[i] - mu) * inv * gam[d] + bet[d] + xbuf[i];
}

__global__ void k_final(const float* __restrict__ outb, const float* __restrict__ bias,
                        const float* __restrict__ xbuf, float* __restrict__ out) {
  int i = blockIdx.x * 256 + threadIdx.x;
  int d = i & (DD - 1);
  out[i] = outb[i] + bias[d] + xbuf[i];
}

// ---------------- host launch ----------------

extern "C" void kernel_launch(void* const* d_in, const int* in_sizes, int n_in,
                              void* d_out, int out_size, void* d_ws, size_t ws_size,
                              hipStream_t stream) {
  const float* X_n   = (const float*)d_in[0];
  const float* PE    = (const float*)d_in[1];
  const int*   eidx  = (const int*)d_in[2];
  const int*   eattr = (const int*)d_in[3];
  const float* eemb  = (const float*)d_in[4];
  const float* peW   = (const float*)d_in[5];
  const float* peB   = (const float*)d_in[6];
  const float* linW  = (const float*)d_in[7];
  const float* attl  = (const float*)d_in[8];
  const float* attr_ = (const float*)d_in[9];
  const float* atte  = (const float*)d_in[10];
  const float* bias  = (const float*)d_in[11];
  const float* gam   = (const float*)d_in[12];
  const float* bet   = (const float*)d_in[13];
  const int* src = eidx;
  const int* dst = eidx + EE;

  float* ws = (float*)d_ws;
  size_t o = 0;
  float* xbuf  = ws + o; o += (size_t)NN * DD;
  float* tmp   = ws + o; o += (size_t)NN * DD;   // aliased: also aggregation output
  float* h     = ws + o; o += (size_t)NN * DD;
  float* PEp   = ws + o; o += (size_t)NN * PEDP;
  float* Wp    = ws + o; o += (size_t)PEDP * DD;
  float* sl    = ws + o; o += NN;
  float* sr    = ws + o; o += NN;
  int*   amax  = (int*)(ws + o); o += NN;
  float* denom = ws + o; o += NN;
  float* alpha = ws + o; o += EE;
  float* xe    = ws + o; o += 16;
  float* ssum  = ws + o; o += DD;
  float* ssq   = ws + o; o += DD;
  float* outb  = tmp;   // tmp is dead after k_gemm<DD>, reuse for aggregation

  const int ND_BLKS = (NN * DD) / 256;  // 25000

  k_copy_x<<<ND_BLKS, 256, 0, stream>>>(X_n, xbuf);
  k_pad_pe<<<(NN * PEDP + 255) / 256, 256, 0, stream>>>(PE, PEp);

  for (int l = 0; l < NL; ++l) {
    k_pad_w<<<(PEDP * DD + 255) / 256, 256, 0, stream>>>(peW + (size_t)l * PED * DD, Wp);
    // tmp = x + PEp @ Wp + pe_b[l]
    k_gemm<PEDP, true><<<3125, 256, 0, stream>>>(PEp, Wp, xbuf, peB + l * DD, tmp);
    // h = tmp @ lin_W[l]
    k_gemm<DD, false><<<3125, 256, 0, stream>>>(tmp, linW + (size_t)l * DD * DD,
                                                nullptr, nullptr, h);
    // zero aggregation output + softmax state (tmp no longer needed as GEMM input)
    k_init<<<ND_BLKS, 256, 0, stream>>>(outb, amax, denom, ssum, ssq);

    k_xe<<<1, 288, 0, stream>>>(eemb + (size_t)l * NT * DD, atte + l * DD, xe);
    k_scores<<<NN / 8, 256, 0, stream>>>(h, attl + l * DD, attr_ + l * DD, sl, sr);

    k_edge_max<<<EE / 256, 256, 0, stream>>>(sl, sr, xe, src, dst, eattr, alpha, amax);
    k_edge_exp<<<EE / 256, 256, 0, stream>>>(amax, dst, alpha, denom);
    k_aggregate<<<EE / 8, 256, 0, stream>>>(h, alpha, denom, src, dst, outb);

    if (l < NL - 1) {
      k_bias_stats<<<(NN + 127) / 128, 128, 0, stream>>>(outb, bias + l * DD, ssum, ssq);
      k_bn_residual<<<ND_BLKS, 256, 0, stream>>>(outb, ssum, ssq, gam + l * DD,
                                                 bet + l * DD, xbuf);
    } else {
      k_final<<<ND_BLKS, 256, 0, stream>>>(outb, bias + l * DD, xbuf, (float*)d_out);
    }
  }
}